// Sampler_66529043415408
// MI455X (gfx1250) — compile-verified
//
#include <hip/hip_runtime.h>
#include <hip/hip_bf16.h>
#include <stdint.h>

// ---------------- problem constants ----------------
#define B_ROWS   256
#define VOCAB    128000
#define SPLIT    5                      // column chunks per row
#define CHUNK    (VOCAB / SPLIT)        // 25600
#define TPB      256                    // 8 waves (wave32)
#define ITER     (CHUNK / (TPB * 4))    // 25 exact float4 tiles per thread
#define NSTAGE   4                      // async pipeline depth (power of 2)
#define NOISE_MIN 1e-10f
#define LN2      0.69314718055994530942f

typedef float v4f __attribute__((ext_vector_type(4)));
typedef int   v4i __attribute__((ext_vector_type(4)));

// ---------------- CDNA5 async global->LDS path (probe-guarded) ----------------
#if __has_builtin(__builtin_amdgcn_global_load_async_to_lds_b128)
#define USE_ASYNC 1
#else
#define USE_ASYNC 0
#endif

#if __has_builtin(__builtin_amdgcn_s_wait_asynccnt)
#define WAIT_ASYNC(n) __builtin_amdgcn_s_wait_asynccnt(n)
#else
#define WAIT_ASYNC(n) asm volatile("s_wait_asynccnt %0" ::"n"(n) : "memory")
#endif

#if USE_ASYNC
__device__ __forceinline__ void async_copy16(const float* g, float* lds) {
  __builtin_amdgcn_global_load_async_to_lds_b128(
      (__attribute__((address_space(1))) v4i*)(g),
      (__attribute__((address_space(3))) v4i*)(lds),
      /*offset=*/0, /*cpol=*/0);
}
#endif

// ---------------- helpers ----------------
__device__ __forceinline__ void upd(float v, int idx, float& bv, int& bi) {
  if (v > bv) { bv = v; bi = idx; }
}

// monotone float->u32 map; key orders by value, ties -> smallest index wins
__device__ __forceinline__ unsigned long long pack_key(float v, int idx) {
  unsigned int u = __float_as_uint(v);
  u = (u & 0x80000000u) ? ~u : (u | 0x80000000u);
  return ((unsigned long long)u << 32) |
         (unsigned long long)(0xFFFFFFFFu - (unsigned int)idx);
}

// ---------------- kernels ----------------
__global__ void init_ws_k(unsigned long long* __restrict__ ws) {
  ws[threadIdx.x] = 0ull;
}

__global__ __launch_bounds__(TPB) void sampler_partial_k(
    const float* __restrict__ logits, const float* __restrict__ temps,
    const float* __restrict__ noise, unsigned long long* __restrict__ ws) {
  const int tid     = threadIdx.x;
  const int row     = blockIdx.y;
  const int colBase = blockIdx.x * CHUNK;

  const float t      = temps[row];        // uniform -> s_load
  const bool  greedy = (t <= 0.0f);
  const float tln2   = t * LN2;           // key = logit - tln2 * log2(noise')

  const float* lg = logits + (size_t)row * VOCAB + colBase;
  const float* nz = noise  + (size_t)row * VOCAB + colBase;

  float bv = -__builtin_inff();
  int   bi = 0;

#if USE_ASYNC
  __shared__ float sL[NSTAGE][TPB * 4];
  __shared__ float sN[NSTAGE][TPB * 4];
  const int so = tid * 4;

  if (greedy) {
#pragma unroll
    for (int s = 0; s < NSTAGE; ++s)
      async_copy16(lg + (s * TPB + tid) * 4, &sL[s][so]);
    for (int i = 0; i < ITER - NSTAGE; ++i) {
      WAIT_ASYNC(NSTAGE - 1);                       // stage i complete
      const int s = i & (NSTAGE - 1);
      v4f L = *(const v4f*)(&sL[s][so]);
      const int off = colBase + (i * TPB + tid) * 4;
      upd(L.x, off + 0, bv, bi); upd(L.y, off + 1, bv, bi);
      upd(L.z, off + 2, bv, bi); upd(L.w, off + 3, bv, bi);
      async_copy16(lg + ((i + NSTAGE) * TPB + tid) * 4, &sL[s][so]);
    }
    WAIT_ASYNC(0);
    for (int i = ITER - NSTAGE; i < ITER; ++i) {
      const int s = i & (NSTAGE - 1);
      v4f L = *(const v4f*)(&sL[s][so]);
      const int off = colBase + (i * TPB + tid) * 4;
      upd(L.x, off + 0, bv, bi); upd(L.y, off + 1, bv, bi);
      upd(L.z, off + 2, bv, bi); upd(L.w, off + 3, bv, bi);
    }
  } else {
#pragma unroll
    for (int s = 0; s < NSTAGE; ++s) {
      async_copy16(lg + (s * TPB + tid) * 4, &sL[s][so]);
      async_copy16(nz + (s * TPB + tid) * 4, &sN[s][so]);
    }
    for (int i = 0; i < ITER - NSTAGE; ++i) {
      WAIT_ASYNC(2 * (NSTAGE - 1));                 // both copies of stage i done
      const int s = i & (NSTAGE - 1);
      v4f L = *(const v4f*)(&sL[s][so]);
      v4f N = *(const v4f*)(&sN[s][so]);
      const int off = colBase + (i * TPB + tid) * 4;
      upd(__builtin_fmaf(-tln2, __builtin_amdgcn_logf(fmaxf(N.x, NOISE_MIN)), L.x), off + 0, bv, bi);
      upd(__builtin_fmaf(-tln2, __builtin_amdgcn_logf(fmaxf(N.y, NOISE_MIN)), L.y), off + 1, bv, bi);
      upd(__builtin_fmaf(-tln2, __builtin_amdgcn_logf(fmaxf(N.z, NOISE_MIN)), L.z), off + 2, bv, bi);
      upd(__builtin_fmaf(-tln2, __builtin_amdgcn_logf(fmaxf(N.w, NOISE_MIN)), L.w), off + 3, bv, bi);
      async_copy16(lg + ((i + NSTAGE) * TPB + tid) * 4, &sL[s][so]);
      async_copy16(nz + ((i + NSTAGE) * TPB + tid) * 4, &sN[s][so]);
    }
    WAIT_ASYNC(0);
    for (int i = ITER - NSTAGE; i < ITER; ++i) {
      const int s = i & (NSTAGE - 1);
      v4f L = *(const v4f*)(&sL[s][so]);
      v4f N = *(const v4f*)(&sN[s][so]);
      const int off = colBase + (i * TPB + tid) * 4;
      upd(__builtin_fmaf(-tln2, __builtin_amdgcn_logf(fmaxf(N.x, NOISE_MIN)), L.x), off + 0, bv, bi);
      upd(__builtin_fmaf(-tln2, __builtin_amdgcn_logf(fmaxf(N.y, NOISE_MIN)), L.y), off + 1, bv, bi);
      upd(__builtin_fmaf(-tln2, __builtin_amdgcn_logf(fmaxf(N.z, NOISE_MIN)), L.z), off + 2, bv, bi);
      upd(__builtin_fmaf(-tln2, __builtin_amdgcn_logf(fmaxf(N.w, NOISE_MIN)), L.w), off + 3, bv, bi);
    }
  }
#else
  // Fallback: direct non-temporal b128 streaming, unrolled x5 for MLP.
  if (greedy) {
    for (int i0 = 0; i0 < ITER; i0 += 5) {
      v4f L[5];
#pragma unroll
      for (int j = 0; j < 5; ++j)
        L[j] = __builtin_nontemporal_load((const v4f*)(lg + ((i0 + j) * TPB + tid) * 4));
#pragma unroll
      for (int j = 0; j < 5; ++j) {
        const int off = colBase + ((i0 + j) * TPB + tid) * 4;
        upd(L[j].x, off + 0, bv, bi); upd(L[j].y, off + 1, bv, bi);
        upd(L[j].z, off + 2, bv, bi); upd(L[j].w, off + 3, bv, bi);
      }
    }
  } else {
    for (int i0 = 0; i0 < ITER; i0 += 5) {
      v4f L[5], N[5];
#pragma unroll
      for (int j = 0; j < 5; ++j) {
        const int e = ((i0 + j) * TPB + tid) * 4;
        L[j] = __builtin_nontemporal_load((const v4f*)(lg + e));
        N[j] = __builtin_nontemporal_load((const v4f*)(nz + e));
      }
#pragma unroll
      for (int j = 0; j < 5; ++j) {
        const int off = colBase + ((i0 + j) * TPB + tid) * 4;
        upd(__builtin_fmaf(-tln2, __builtin_amdgcn_logf(fmaxf(N[j].x, NOISE_MIN)), L[j].x), off + 0, bv, bi);
        upd(__builtin_fmaf(-tln2, __builtin_amdgcn_logf(fmaxf(N[j].y, NOISE_MIN)), L[j].y), off + 1, bv, bi);
        upd(__builtin_fmaf(-tln2, __builtin_amdgcn_logf(fmaxf(N[j].z, NOISE_MIN)), L[j].z), off + 2, bv, bi);
        upd(__builtin_fmaf(-tln2, __builtin_amdgcn_logf(fmaxf(N[j].w, NOISE_MIN)), L[j].w), off + 3, bv, bi);
      }
    }
  }
#endif

  // ---- block reduction over packed (value,index) keys in LDS ----
  __shared__ unsigned long long red[TPB];
  red[tid] = pack_key(bv, bi);
  __syncthreads();
#pragma unroll
  for (int s = TPB / 2; s > 0; s >>= 1) {
    if (tid < s) {
      unsigned long long o = red[tid + s];
      if (o > red[tid]) red[tid] = o;
    }
    __syncthreads();
  }
  if (tid == 0) atomicMax(&ws[row], red[0]);   // global_atomic_max_u64
}

__global__ void finalize_k(const unsigned long long* __restrict__ ws,
                           long long* __restrict__ out) {
  const int i = threadIdx.x;
  const unsigned int idx = 0xFFFFFFFFu - (unsigned int)(ws[i] & 0xFFFFFFFFull);
  out[i] = (long long)idx;
}

// ---------------- launch ----------------
extern "C" void kernel_launch(void* const* d_in, const int* in_sizes, int n_in,
                              void* d_out, int out_size, void* d_ws, size_t ws_size,
                              hipStream_t stream) {
  const float* logits = (const float*)d_in[0];   // [B, V] f32
  const float* temps  = (const float*)d_in[1];   // [B]    f32
  const float* noise  = (const float*)d_in[2];   // [B, V] f32
  unsigned long long* ws = (unsigned long long*)d_ws;   // B u64 partial keys
  long long* out = (long long*)d_out;                   // [B] int64

  hipLaunchKernelGGL(init_ws_k, dim3(1), dim3(B_ROWS), 0, stream, ws);
  hipLaunchKernelGGL(sampler_partial_k, dim3(SPLIT, B_ROWS), dim3(TPB), 0, stream,
                     logits, temps, noise, ws);
  hipLaunchKernelGGL(finalize_k, dim3(1), dim3(B_ROWS), 0, stream, ws, out);
}